// GCN_21895743275233
// MI455X (gfx1250) — compile-verified
//
#include <hip/hip_runtime.h>

// ---------------------------------------------------------------------------
// GCN on MI455X (gfx1250, wave32).
// GEMMs use V_WMMA_F32_16X16X4_F32 (full fp32, matches reference precision;
// GEMM flops ~23G are negligible vs the L2-resident SpMM scatter traffic).
// B (weight) tiles are staged in LDS once per block and read as conflict-free
// ds_load_b64 fragments; A rows stream as global b64 loads.
// SpMM: float4 gather + global_atomic_add_f32 scatter; feature matrices
// (51.2 MB) are L2-resident on the 192 MB L2.
// ---------------------------------------------------------------------------

typedef float v2f __attribute__((ext_vector_type(2)));
typedef float v8f __attribute__((ext_vector_type(8)));

#define N_NODES 50000
#define N_EDGES 1600000
#define NFEAT   512
#define NHID    256
#define NOUT    128
#define NCLS    16

#define KCHUNK   32            // K rows staged per LDS chunk
#define LDS_PAD  34            // column stride in floats (even, 16 distinct banks)

// C[N x Fout] = A[N x Fin] * W[Fin x Fout], row-major everywhere.
// Block = 8 waves: 8 row-tiles (128 rows) x one 64-col group; all waves share
// the same B fragments, staged once per block into LDS.
// A-frag (16x4 f32): lane L, vgpr v holds A[L%16][k + 2*(L>>4) + v]  (b64 load)
// B-frag (4x16 f32): lane L, vgpr v holds W[k + 2*(L>>4) + v][L%16]  (ds b64)
// C/D    (16x16 f32): vgpr r, lane L holds C[r + 8*(L>>4)][L%16]
__global__ void __launch_bounds__(256)
gemm_wmma_f32(const float* __restrict__ A, const float* __restrict__ W,
              float* __restrict__ C, int N, int Fin, int Fout) {
  __shared__ float tile[64 * LDS_PAD];             // tile[c*LDS_PAD + k]

  const int lane = threadIdx.x & 31;
  const int waveId = threadIdx.x >> 5;
  const int colGroups = Fout >> 6;
  const int blockRow = blockIdx.x / colGroups;     // in units of 8 row-tiles
  const int colGroup = blockIdx.x - blockRow * colGroups;
  const int rowTile = blockRow * 8 + waveId;
  const bool valid = (rowTile * 16) < N;
  const int half = lane >> 4;
  const int l16  = lane & 15;

  // Invalid waves clamp to row 0 (safe loads) but still hit every barrier.
  const int arowIdx = valid ? (rowTile * 16 + l16) : l16;
  const float* __restrict__ arow = A + (size_t)arowIdx * Fin;
  const float* __restrict__ wblkBase = W + colGroup * 64;

  v8f acc0 = {}, acc1 = {}, acc2 = {}, acc3 = {};
  for (int k0 = 0; k0 < Fin; k0 += KCHUNK) {
    __syncthreads();                               // protect LDS reuse
    // Stage W[k0..k0+31][colGroup*64 .. +64] transposed: tile[c][k] layout.
    // 2048 floats / 256 threads = 8 each; consecutive threads -> consecutive
    // cols of the same k-row (coalesced 256B global reads).
    const float* __restrict__ wblk = wblkBase + (size_t)k0 * Fout;
#pragma unroll
    for (int i = 0; i < 8; ++i) {
      int idx = threadIdx.x + 256 * i;             // 0..2047
      int r = idx >> 6;                            // k offset 0..31
      int c = idx & 63;                            // col    0..63
      tile[c * LDS_PAD + r] = wblk[(size_t)r * Fout + c];
    }
    __syncthreads();

#pragma unroll
    for (int kk = 0; kk < KCHUNK; kk += 4) {
      v2f a = *(const v2f*)(arow + k0 + kk + 2 * half);
      const float* bp = &tile[l16 * LDS_PAD + kk + 2 * half];
      v2f b0 = *(const v2f*)(bp);
      v2f b1 = *(const v2f*)(bp + 16 * LDS_PAD);
      v2f b2 = *(const v2f*)(bp + 32 * LDS_PAD);
      v2f b3 = *(const v2f*)(bp + 48 * LDS_PAD);
      acc0 = __builtin_amdgcn_wmma_f32_16x16x4_f32(false, a, false, b0, (short)0, acc0, false, false);
      acc1 = __builtin_amdgcn_wmma_f32_16x16x4_f32(false, a, false, b1, (short)0, acc1, false, false);
      acc2 = __builtin_amdgcn_wmma_f32_16x16x4_f32(false, a, false, b2, (short)0, acc2, false, false);
      acc3 = __builtin_amdgcn_wmma_f32_16x16x4_f32(false, a, false, b3, (short)0, acc3, false, false);
    }
  }

  if (valid) {
#pragma unroll
    for (int r = 0; r < 8; ++r) {
      float* crow = C + (size_t)(rowTile * 16 + r + 8 * half) * Fout + colGroup * 64 + l16;
      crow[0]  = acc0[r];
      crow[16] = acc1[r];
      crow[32] = acc2[r];
      crow[48] = acc3[r];
    }
  }
}

__global__ void zero_f32(float* __restrict__ p, long long n) {
  long long i = (long long)blockIdx.x * blockDim.x + threadIdx.x;
  long long stride = (long long)gridDim.x * blockDim.x;
  for (; i < n; i += stride) p[i] = 0.0f;
}

// agg[dst[e]] += w[e] * support[src[e]] ; one float4 chunk per lane.
// chunks (64 or 32) is a multiple of 32, so each wave maps to exactly one
// edge: index loads are wave-uniform, gather/atomics are 512B coalesced.
__global__ void __launch_bounds__(256)
spmm_scatter(const float* __restrict__ support, const int* __restrict__ src,
             const int* __restrict__ dst, const float* __restrict__ ew,
             float* __restrict__ agg, int E, int F) {
  int tid = blockIdx.x * blockDim.x + threadIdx.x;
  int chunks = F >> 2;
  int e = tid / chunks;
  if (e >= E) return;
  int c = tid - e * chunks;
  int s = src[e], d = dst[e];
  float w = ew[e];
  const float4 v = *(const float4*)(support + (size_t)s * F + c * 4);
  float* o = agg + (size_t)d * F + c * 4;
  atomicAdd(o + 0, w * v.x);
  atomicAdd(o + 1, w * v.y);
  atomicAdd(o + 2, w * v.z);
  atomicAdd(o + 3, w * v.w);
}

__global__ void bias_act(float* __restrict__ h, const float* __restrict__ b,
                         int total, int F, int do_relu) {
  int tid = blockIdx.x * blockDim.x + threadIdx.x;
  if (tid >= total) return;
  int f = tid % F;
  float v = h[tid] + b[f];
  h[tid] = do_relu ? fmaxf(v, 0.0f) : v;
}

// logits = concat(x1,x2,x3) @ linW + linb, then log_softmax over 16 classes.
// One wave handles 2 nodes: lanes 0-15 -> node a, lanes 16-31 -> node b;
// lane%16 = class index; shuffle reductions with width 16 stay in-half.
__global__ void __launch_bounds__(256)
final_logits(const float* __restrict__ x1, const float* __restrict__ x2,
             const float* __restrict__ x3, const float* __restrict__ linW,
             const float* __restrict__ linb, float* __restrict__ out, int N) {
  int wave = blockIdx.x * (blockDim.x >> 5) + (threadIdx.x >> 5);
  int lane = threadIdx.x & 31;
  int half = lane >> 4;
  int c = lane & 15;
  int n = wave * 2 + half;
  if (n >= N) return;

  float acc = linb[c];
  const float* p1 = x1 + (size_t)n * NHID;
  for (int f = 0; f < NHID; ++f) acc += p1[f] * linW[f * NCLS + c];
  const float* p2 = x2 + (size_t)n * NHID;
  for (int f = 0; f < NHID; ++f) acc += p2[f] * linW[(NHID + f) * NCLS + c];
  const float* p3 = x3 + (size_t)n * NOUT;
  for (int f = 0; f < NOUT; ++f) acc += p3[f] * linW[(2 * NHID + f) * NCLS + c];

  float m = acc;
  for (int off = 8; off; off >>= 1) m = fmaxf(m, __shfl_xor(m, off, 16));
  float ex = __expf(acc - m);
  float s = ex;
  for (int off = 8; off; off >>= 1) s += __shfl_xor(s, off, 16);
  out[(size_t)n * NCLS + c] = acc - m - __logf(s);
}

extern "C" void kernel_launch(void* const* d_in, const int* in_sizes, int n_in,
                              void* d_out, int out_size, void* d_ws, size_t ws_size,
                              hipStream_t stream) {
  const float* x    = (const float*)d_in[0];
  const float* ew   = (const float*)d_in[1];
  const float* W1   = (const float*)d_in[2];
  const float* b1   = (const float*)d_in[3];
  const float* W2   = (const float*)d_in[4];
  const float* b2   = (const float*)d_in[5];
  const float* W3   = (const float*)d_in[6];
  const float* b3   = (const float*)d_in[7];
  const float* linW = (const float*)d_in[8];
  const float* linb = (const float*)d_in[9];
  const int* esrc   = (const int*)d_in[10];
  const int* edst   = (const int*)d_in[11];
  float* out = (float*)d_out;

  // workspace layout: support | x1 | x2 | x3   (~179 MB total)
  float* support = (float*)d_ws;
  float* x1 = support + (size_t)N_NODES * NHID;
  float* x2 = x1 + (size_t)N_NODES * NHID;
  float* x3 = x2 + (size_t)N_NODES * NHID;

  auto gemm = [&](const float* Ain, const float* Wm, float* Cout, int Fin, int Fout) {
    int rowTiles  = (N_NODES + 15) / 16;
    int rowBlocks = (rowTiles + 7) / 8;
    int blocks    = rowBlocks * (Fout / 64);
    gemm_wmma_f32<<<blocks, 256, 0, stream>>>(Ain, Wm, Cout, N_NODES, Fin, Fout);
  };
  auto zero = [&](float* p, long long n) {
    zero_f32<<<2048, 256, 0, stream>>>(p, n);
  };
  auto spmm = [&](const float* sup, float* agg, int F) {
    long long total = (long long)N_EDGES * (F / 4);
    int blocks = (int)((total + 255) / 256);
    spmm_scatter<<<blocks, 256, 0, stream>>>(sup, esrc, edst, ew, agg, N_EDGES, F);
  };
  auto bias = [&](float* h, const float* b, int F, int relu) {
    int total = N_NODES * F;
    bias_act<<<(total + 255) / 256, 256, 0, stream>>>(h, b, total, F, relu);
  };

  // Layer 1: x -> x1 (512 -> 256, ReLU)
  gemm(x, W1, support, NFEAT, NHID);
  zero(x1, (long long)N_NODES * NHID);
  spmm(support, x1, NHID);
  bias(x1, b1, NHID, 1);

  // Layer 2: x1 -> x2 (256 -> 256, ReLU)
  gemm(x1, W2, support, NHID, NHID);
  zero(x2, (long long)N_NODES * NHID);
  spmm(support, x2, NHID);
  bias(x2, b2, NHID, 1);

  // Layer 3: x2 -> x3 (256 -> 128, no ReLU)
  gemm(x2, W3, support, NHID, NOUT);
  zero(x3, (long long)N_NODES * NOUT);
  spmm(support, x3, NOUT);
  bias(x3, b3, NOUT, 0);

  // Final linear + log_softmax
  int fwaves  = (N_NODES + 1) / 2;
  int fblocks = (fwaves * 32 + 255) / 256;
  final_logits<<<fblocks, 256, 0, stream>>>(x1, x2, x3, linW, linb, out, N_NODES);
}